// Conv2dGeneral_10170482557516
// MI455X (gfx1250) — compile-verified
//
#include <hip/hip_runtime.h>

// Capsule-conv as a single GEMM on the CDNA5 fp32 WMMA path:
//   D[M,N] = A[M,K] * B[K,N]
//   M = Cout*4 = 128  rows   (o,u)
//   N = B*13*13*4 = 43264    (b,x,y,w)
//   K = Cin*3*3*4 = 1152     (p,v)
//   A[(o,u),(p,v)] = W[p*512 + (o*4+u)*4 + v]        (v contiguous -> b64 loads)
//   B[(p,v),(b,x,y,w)] = x[base(b,x,y,p) + v*4 + w]  (staged via LDS, transposed
//                                                     to [bxy][w][v] so each lane's
//                                                     K pair is one ds_load_b64)
// Each wave computes a 16x32 D region (two 16x16 tiles) so every A-fragment
// b64 load feeds two v_wmma; the 8 waves of a block share one LDS x tile.

typedef float v2f __attribute__((ext_vector_type(2)));
typedef float v8f __attribute__((ext_vector_type(8)));

#define B_    64
#define CIN   32
#define COUT  32
#define KSZ   3
#define STR   2
#define WIN   28
#define HH    16
#define WOUT  13
#define P_TOT (CIN * KSZ * KSZ)                 // 288
#define PCHUNK 16                               // p's staged per LDS round
#define NBLK  ((B_ * WOUT * WOUT * 4) / 32)     // 1352 blocks, 32 N-cols each

__global__ __launch_bounds__(256)
void capsule_conv_wmma(const float* __restrict__ x,
                       const float* __restrict__ W,
                       const float* __restrict__ bias,
                       float* __restrict__ out)
{
    // LDS stage: PCHUNK p x 8 (b,x,y) x 16 floats (layout [bxy][w][v]) = 8 KB
    __shared__ float lds[PCHUNK * 128];

    const int tid  = threadIdx.x;
    const int wave = tid >> 5;          // 0..7 : M-tile index
    const int lane = tid & 31;
    const int hi   = lane >> 4;         // lane half selects K pair (v = 2*hi + j)
    const int lo   = lane & 15;

    const int nblk = blockIdx.x;        // 0..1351 : 32 N columns per block

    // Per-lane output columns for the two N-tiles
    const int n0 = nblk * 32 + lo;
    const int n1 = n0 + 16;
    const int w0 = n0 & 3, w1 = n1 & 3;
    const int bxy0 = n0 >> 2, bxy1 = n1 >> 2;
    const int b0 = bxy0 / (WOUT * WOUT), b1 = bxy1 / (WOUT * WOUT);
    const int r0 = bxy0 % (WOUT * WOUT), r1 = bxy1 % (WOUT * WOUT);
    const int xq0 = r0 / WOUT, xq1 = r1 / WOUT;
    const int yq0 = r0 % WOUT, yq1 = r1 % WOUT;

    const int mrow = wave * 16 + lo;     // A-matrix row (o*4 + u)
    const int v0   = 2 * hi;

    v8f acc0 = {0.f,0.f,0.f,0.f,0.f,0.f,0.f,0.f};
    v8f acc1 = {0.f,0.f,0.f,0.f,0.f,0.f,0.f,0.f};

    for (int pc = 0; pc < P_TOT; pc += PCHUNK) {
        // ---- cooperative stage of x tile into LDS ----
        // 512 float4 segments/chunk; thread handles f = tid and tid+256.
        // f -> p_local = f>>5, bxy_local = (f>>2)&7, v = f&3.
        // Global read: 4 contiguous floats (w=0..3) for that v; LDS write is
        // a stride-4 scatter into [bxy][w][v] (transposed for b64 reads).
        #pragma unroll
        for (int half = 0; half < 2; ++half) {
            const int f     = tid + half * 256;
            const int sp_p  = f >> 5;            // 0..15
            const int sp_bl = (f >> 2) & 7;      // 0..7 bxy within block
            const int sp_v  = f & 3;             // pose column v

            const int p   = pc + sp_p;
            const int cin = p / (KSZ * KSZ);
            const int kr  = p % (KSZ * KSZ);
            const int ki  = kr / KSZ;
            const int kj  = kr % KSZ;

            const int s_bxy = nblk * 8 + sp_bl;
            const int s_b   = s_bxy / (WOUT * WOUT);
            const int s_rem = s_bxy % (WOUT * WOUT);
            const int s_x   = s_rem / WOUT;
            const int s_y   = s_rem % WOUT;

            const float* src = x +
                ((((size_t)s_b * CIN + cin) * WIN + (STR * s_x + ki)) * WIN
                 + (STR * s_y + kj)) * HH + sp_v * 4;
            float4 v4 = *(const float4*)src;     // w = 0..3 for this v
            float* dst = &lds[sp_p * 128 + sp_bl * 16 + sp_v];
            dst[0]  = v4.x;   // w=0
            dst[4]  = v4.y;   // w=1
            dst[8]  = v4.z;   // w=2
            dst[12] = v4.w;   // w=3
        }
        __syncthreads();

        // ---- 16 K-steps: 1 global b64 (A) + 2 ds b64 (B) + 2 wmma each ----
        const float* wbase = W + (size_t)pc * (COUT * HH) + mrow * 4 + v0;
        #pragma unroll
        for (int pp = 0; pp < PCHUNK; ++pp) {
            v2f a = *(const v2f*)(wbase + pp * (COUT * HH));

            v2f bf0 = *(const v2f*)&lds[pp * 128 + (lo >> 2) * 16 + w0 * 4 + v0];
            v2f bf1 = *(const v2f*)&lds[pp * 128 + 64 + (lo >> 2) * 16 + w1 * 4 + v0];

            acc0 = __builtin_amdgcn_wmma_f32_16x16x4_f32(
                false, a, false, bf0, (short)0, acc0, false, false);
            acc1 = __builtin_amdgcn_wmma_f32_16x16x4_f32(
                false, a, false, bf1, (short)0, acc1, false, false);
        }
        __syncthreads();
    }

    // ---- bias + store: D VGPR r holds M = wave*16 + 8*hi + r ----
    #pragma unroll
    for (int r = 0; r < 8; ++r) {
        const int m = wave * 16 + 8 * hi + r;
        const int o = m >> 2;
        const int u = m & 3;
        const float bv = bias[o];
        const size_t oi0 =
            ((((size_t)b0 * COUT + o) * WOUT + xq0) * WOUT + yq0) * HH + u * 4 + w0;
        const size_t oi1 =
            ((((size_t)b1 * COUT + o) * WOUT + xq1) * WOUT + yq1) * HH + u * 4 + w1;
        out[oi0] = acc0[r] + bv;
        out[oi1] = acc1[r] + bv;
    }
}

extern "C" void kernel_launch(void* const* d_in, const int* in_sizes, int n_in,
                              void* d_out, int out_size, void* d_ws, size_t ws_size,
                              hipStream_t stream) {
    const float* x    = (const float*)d_in[0];
    const float* W    = (const float*)d_in[1];
    const float* bias = (const float*)d_in[2];
    float* out = (float*)d_out;
    dim3 grid(NBLK), block(256);
    hipLaunchKernelGGL(capsule_conv_wmma, grid, block, 0, stream, x, W, bias, out);
}